// SentenceContrastive_13314398618441
// MI455X (gfx1250) — compile-verified
//
#include <hip/hip_runtime.h>
#include <math.h>

// Problem constants (from reference): B=256, m=128, d=768, TEMP=0.07
#define B_SZ   256
#define M_TOK  128
#define D_DIM  768
#define TEMP_INV (1.0f / 0.07f)

typedef __attribute__((ext_vector_type(2))) float v2f;
typedef __attribute__((ext_vector_type(8))) float v8f;

// ---------------------------------------------------------------------------
// Kernel 1: sim_max[i][j] = (1/TEMP) * max_m dot(v_final[i,:], T_fused[j,m,:])
//
// grid = (j: 256, i_block: 2), block = 256 threads (8 waves).
// Wave w handles text-token chunk m in [16w, 16w+16) and accumulates
// NT=8 WMMA C-tiles covering i in [i_block*128, i_block*128+128).
// Each K-step of 4 uses V_WMMA_F32_16X16X4_F32 (fp32 in/out, full precision).
//
// Fragment layouts (CDNA5 ISA 7.12.2, 32-bit A 16x4 / B 4x16):
//   lane l: n = l&15 (row M for A, col N for B), koff = (l>>4)*2
//   A frag = { v_final[i0 + n][k+koff], v_final[i0 + n][k+koff+1] }
//   B frag = { T_fused[j][m0 + n][k+koff], T_fused[j][m0 + n][k+koff+1] }
// C layout: VGPR r -> (M=r lanes 0..15, M=r+8 lanes 16..31), N = lane&15.
// ---------------------------------------------------------------------------
#define NT 8  // i-tiles (of 16 rows) per wave

__global__ __launch_bounds__(256)
void simmax_wmma_kernel(const float* __restrict__ v_final,
                        const float* __restrict__ T_fused,
                        float* __restrict__ sim_max) {
    const int j      = blockIdx.x;
    const int i_base = blockIdx.y * (NT * 16);   // 0 or 128
    const int lane   = threadIdx.x & 31;
    const int wave   = threadIdx.x >> 5;
    const int n      = lane & 15;
    const int koff   = (lane >> 4) * 2;
    const int m0     = wave * 16;

    __shared__ float wave_max[8][NT * 16];       // per-wave row maxima

    const float* bptr = T_fused + ((size_t)j * M_TOK + m0 + n) * D_DIM + koff;
    const float* aptr = v_final + (size_t)(i_base + n) * D_DIM + koff;

    v8f c[NT];
    #pragma unroll
    for (int t = 0; t < NT; ++t) {
        v8f z = {};
        c[t] = z;
    }

    for (int k = 0; k < D_DIM; k += 4) {
        v2f b = *(const v2f*)(bptr + k);
        #pragma unroll
        for (int t = 0; t < NT; ++t) {
            v2f a = *(const v2f*)(aptr + (size_t)t * 16 * D_DIM + k);
            // D = A(16x4) * B(4x16) + C, fp32 throughout
            c[t] = __builtin_amdgcn_wmma_f32_16x16x4_f32(
                /*neg_a=*/false, a, /*neg_b=*/false, b,
                /*c_mod=*/(short)0, c[t], /*reuse_a=*/false, /*reuse_b=*/false);
        }
    }

    // Max over N (the 16 tokens of this wave's chunk): reduce across each
    // 16-lane half; lanes 0 and 16 then hold the maxima for M=r and M=r+8.
    #pragma unroll
    for (int t = 0; t < NT; ++t) {
        #pragma unroll
        for (int r = 0; r < 8; ++r) {
            float v = c[t][r];
            v = fmaxf(v, __shfl_xor(v, 1, 32));
            v = fmaxf(v, __shfl_xor(v, 2, 32));
            v = fmaxf(v, __shfl_xor(v, 4, 32));
            v = fmaxf(v, __shfl_xor(v, 8, 32));
            if ((lane & 15) == 0) {
                // lane 0 -> M=r, lane 16 -> M=r+8 within this 16-row tile
                wave_max[wave][t * 16 + (lane >> 4) * 8 + r] = v;
            }
        }
    }
    __syncthreads();

    // Fold the 8 waves (8 token chunks -> max over all 128 tokens) and write.
    if (threadIdx.x < NT * 16) {
        float m = wave_max[0][threadIdx.x];
        #pragma unroll
        for (int w = 1; w < 8; ++w) m = fmaxf(m, wave_max[w][threadIdx.x]);
        const int i = i_base + threadIdx.x;
        sim_max[(size_t)i * B_SZ + j] = m * TEMP_INV;
    }
}

// ---------------------------------------------------------------------------
// Kernel 2: symmetric InfoNCE loss from the 256x256 sim_max matrix.
// One block of 256 threads; thread t owns row t and column t.
// loss = sum_t [ -(S[t,t] - lse_row(t)) - (S[t,t] - lse_col(t)) ] / (2*256)
// ---------------------------------------------------------------------------
__global__ __launch_bounds__(256)
void loss_kernel(const float* __restrict__ S, float* __restrict__ out) {
    const int t = threadIdx.x;

    float rmax = -INFINITY, cmax = -INFINITY;
    for (int q = 0; q < B_SZ; ++q) {
        rmax = fmaxf(rmax, S[(size_t)t * B_SZ + q]);
        cmax = fmaxf(cmax, S[(size_t)q * B_SZ + t]);
    }
    float rsum = 0.0f, csum = 0.0f;
    for (int q = 0; q < B_SZ; ++q) {
        rsum += expf(S[(size_t)t * B_SZ + q] - rmax);
        csum += expf(S[(size_t)q * B_SZ + t] - cmax);
    }
    const float diag = S[(size_t)t * B_SZ + t];
    const float li2t = -(diag - rmax - logf(rsum));
    const float lt2i = -(diag - cmax - logf(csum));

    __shared__ float red[B_SZ];
    red[t] = li2t + lt2i;
    __syncthreads();
    for (int s = B_SZ / 2; s > 0; s >>= 1) {
        if (t < s) red[t] += red[t + s];
        __syncthreads();
    }
    if (t == 0) out[0] = red[0] * (1.0f / (2.0f * (float)B_SZ));
}

// ---------------------------------------------------------------------------
extern "C" void kernel_launch(void* const* d_in, const int* in_sizes, int n_in,
                              void* d_out, int out_size, void* d_ws, size_t ws_size,
                              hipStream_t stream) {
    const float* v_final = (const float*)d_in[0];  // [256, 768] f32
    const float* T_fused = (const float*)d_in[1];  // [256, 128, 768] f32
    float* S = (float*)d_ws;                       // [256, 256] f32 scratch
    float* out = (float*)d_out;                    // scalar f32

    dim3 grid(B_SZ, B_SZ / (NT * 16));             // (256, 2)
    simmax_wmma_kernel<<<grid, 256, 0, stream>>>(v_final, T_fused, S);
    loss_kernel<<<1, B_SZ, 0, stream>>>(S, out);
}